// series_decomp_31533649887832
// MI455X (gfx1250) — compile-verified
//
#include <hip/hip_runtime.h>
#include <math.h>

#define BATCH 32
#define LEN   4096
#define CH    512
#define HALF  12
#define NSEL  40            // padded selection slots (<=33 used)
#define KSEL  80            // 2*NSEL basis functions (cos,-sin per bin)
#define TWO_PI 6.28318530717958647692f

typedef float v2f __attribute__((ext_vector_type(2)));
typedef float v8f __attribute__((ext_vector_type(8)));

// ---- ws layout (float indices) ----
#define ST_OFF   0ull                          // transposed seasonal [B][C][L]
#define AP_OFF   (67108864ull)                 // avg_power 2064
#define FSEL_OFF (AP_OFF + 2064ull)            // 64 ints
#define COEF_OFF (FSEL_OFF + 64ull)            // 64 floats
#define B1_OFF   (COEF_OFF + 64ull)            // [4096][KSEL]
#define BT_OFF   (B1_OFF + 4096ull*KSEL)       // [KSEL][4096]

// ---------------- kernel 0: zero avg_power ----------------
__global__ void k0_init(float* __restrict__ ap) {
  for (int i = threadIdx.x; i < 2064; i += 256) ap[i] = 0.f;
}

// ---------------- kernel A: moving mean + seasonal (transposed) ----------------
__global__ __launch_bounds__(256) void kA_decomp(const float* __restrict__ x,
                                                 float* __restrict__ mm_out,
                                                 float* __restrict__ st) {
  __shared__ float xs[56][33];   // 32 rows + 24 halo, padded
  __shared__ float sv[32][33];   // seasonal tile for transpose
  int c0 = blockIdx.x * 32;
  int n0 = blockIdx.y * 32;
  int b  = blockIdx.z;
  int tx = threadIdx.x & 31;
  int ty = threadIdx.x >> 5;     // 0..7
  const float* xb = x + (size_t)b * LEN * CH;

  for (int r = ty; r < 56; r += 8) {
    int n = n0 + r - HALF;
    n = n < 0 ? 0 : (n > LEN - 1 ? LEN - 1 : n);   // edge replicate
    xs[r][tx] = xb[(size_t)n * CH + c0 + tx];
  }
  __syncthreads();

  float* mmb = mm_out + (size_t)b * LEN * CH;
  #pragma unroll
  for (int k = 0; k < 4; ++k) {
    int r = ty + k * 8;          // 0..31
    float s = 0.f;
    #pragma unroll
    for (int d = 0; d < 25; ++d) s += xs[r + d][tx];
    float mean = s * (1.0f / 25.0f);
    mmb[(size_t)(n0 + r) * CH + c0 + tx] = mean;
    sv[r][tx] = xs[r + HALF][tx] - mean;
  }
  __syncthreads();

  float* stb = st + (size_t)b * CH * LEN;
  #pragma unroll
  for (int k = 0; k < 4; ++k) {
    int c = ty + k * 8;          // 0..31
    stb[(size_t)(c0 + c) * LEN + n0 + tx] = sv[tx][c];
  }
}

// ---------------- kernel B: packed-complex FFT + power spectrum ----------------
__global__ __launch_bounds__(256) void kB_fftpow(const float* __restrict__ st,
                                                 float* __restrict__ ap) {
  __shared__ float2 fbuf[2][4096];   // ping-pong Stockham buffers (64 KB)
  __shared__ float2 tw[4096];        // e^{-2pi i k/4096} (32 KB)
  __shared__ float  pacc[2064];
  int tid = threadIdx.x;

  for (int k = tid; k < 4096; k += 256) {
    float s, c;
    __sincosf((TWO_PI / 4096.0f) * (float)k, &s, &c);
    tw[k] = make_float2(c, -s);
  }
  for (int f = tid; f < 2064; f += 256) pacc[f] = 0.f;
  __syncthreads();

  int pr0 = blockIdx.x * 8;
  for (int pp = 0; pp < 8; ++pp) {
    int pr = pr0 + pp;                 // pair index 0..8191
    int b  = pr >> 8;
    int cp = pr & 255;
    const float* s0 = st + ((size_t)b * CH + 2 * cp) * LEN;
    const float* s1 = s0 + LEN;
    for (int n = tid; n < 4096; n += 256)
      fbuf[0][n] = make_float2(s0[n], s1[n]);    // z = s_even + i*s_odd
    __syncthreads();

    int cur = 0;
    for (int stg = 0; stg < 12; ++stg) {         // Stockham radix-2, self-sorting
      const float2* X = fbuf[cur];
      float2* Y = fbuf[cur ^ 1];
      int m = 2048 >> stg;
      for (int it = tid; it < 2048; it += 256) {
        int q = it & ((1 << stg) - 1);
        int p = it >> stg;
        float2 w  = tw[p << stg];                // e^{-2pi i p / n}
        float2 a  = X[q + (p << stg)];
        float2 bb = X[q + ((p + m) << stg)];
        float2 su = make_float2(a.x + bb.x, a.y + bb.y);
        float2 df = make_float2(a.x - bb.x, a.y - bb.y);
        float2 dw = make_float2(df.x * w.x - df.y * w.y, df.x * w.y + df.y * w.x);
        Y[q + (p << (stg + 1))] = su;
        Y[q + (((p << 1) + 1) << stg)] = dw;
      }
      cur ^= 1;
      __syncthreads();
    }
    const float2* Z = fbuf[cur];                 // cur is back to 0
    for (int f = tid; f <= 2048; f += 256) {     // unpack two real spectra
      float2 zf = Z[f];
      float2 zc = Z[(4096 - f) & 4095];
      float p0 = (zf.x + zc.x) * (zf.x + zc.x) + (zf.y - zc.y) * (zf.y - zc.y);
      float p1 = (zf.y + zc.y) * (zf.y + zc.y) + (zf.x - zc.x) * (zf.x - zc.x);
      pacc[f] += 0.25f * (p0 + p1);
    }
    __syncthreads();
  }
  for (int f = tid; f <= 2048; f += 256) atomicAdd(&ap[f], pacc[f]);
}

// ---------------- kernel C: top-3, selection list, basis tables ----------------
__global__ __launch_bounds__(256) void kC_mask(const float* __restrict__ ap,
                                               int* __restrict__ fsel,
                                               float* __restrict__ coef,
                                               float* __restrict__ B1,
                                               float* __restrict__ Bt) {
  __shared__ float pv[2049];
  __shared__ float rv[256];
  __shared__ int   ri[256];
  __shared__ int   topi[3];
  __shared__ int   fselL[NSEL];
  int tid = threadIdx.x;
  for (int f = tid; f < 2049; f += 256) pv[f] = ap[f];
  __syncthreads();

  for (int k = 0; k < 3; ++k) {
    float bv = -1e38f; int bi = 2049;
    for (int f = tid; f < 2049; f += 256) {
      float v = pv[f];
      if (v > bv || (v == bv && f < bi)) { bv = v; bi = f; }
    }
    rv[tid] = bv; ri[tid] = bi;
    __syncthreads();
    for (int off = 128; off > 0; off >>= 1) {
      if (tid < off) {
        float v2 = rv[tid + off]; int i2 = ri[tid + off];
        if (v2 > rv[tid] || (v2 == rv[tid] && i2 < ri[tid])) { rv[tid] = v2; ri[tid] = i2; }
      }
      __syncthreads();
    }
    if (tid == 0) { topi[k] = ri[0]; pv[ri[0]] = -1e38f; }
    __syncthreads();
  }

  if (tid == 0) {
    int ns = 0;
    for (int f = 0; f < 2049; ++f) {
      float w = 0.f;
      for (int k = 0; k < 3; ++k) {
        int d = f - topi[k]; d = d < 0 ? -d : d;
        if (d <= 5) w += 1.0f - (float)d * (1.0f / 6.0f);
      }
      if (w > 0.f && ns < NSEL) {
        float wsum = 0.5f * w;                      // FILTER_STRENGTH * sum(w)
        float mult = (f == 0 || f == 2048) ? 1.f : 2.f;
        float cf = 0.2f * mult * wsum * (1.0f / 4096.0f);
        fselL[ns] = f; fsel[ns] = f; coef[ns] = cf; ns++;
      }
    }
    for (; ns < NSEL; ++ns) { fselL[ns] = 0; fsel[ns] = 0; coef[ns] = 0.f; }
  }
  __syncthreads();

  for (int idx = tid; idx < 4096 * NSEL; idx += 256) {
    int n = idx / NSEL, j = idx % NSEL;
    int f = fselL[j];
    float s, c;
    __sincosf((TWO_PI / 4096.0f) * (float)((f * n) & 4095), &s, &c);
    B1[(size_t)n * KSEL + 2 * j]     = c;
    B1[(size_t)n * KSEL + 2 * j + 1] = -s;
    Bt[(size_t)(2 * j) * 4096 + n]     = c;
    Bt[(size_t)(2 * j + 1) * 4096 + n] = -s;
  }
}

// ---------------- kernel D: sparse DFT analysis + synthesis via WMMA ----------------
__global__ __launch_bounds__(256) void kD_final(const float* __restrict__ x,
                                                const float* __restrict__ mm,
                                                const float* __restrict__ B1,
                                                const float* __restrict__ Bt,
                                                const float* __restrict__ coef,
                                                float* __restrict__ outF) {
  __shared__ float c2[KSEL][16];        // C2[k][ch] accumulators
  __shared__ float ldsbuf[8 * 80 * 33]; // phase A: Bt tiles; phase B: B1 tiles (union)
  int b  = blockIdx.y;
  int c0 = blockIdx.x * 16;
  int tid  = threadIdx.x;
  int w    = tid >> 5;
  int lane = tid & 31;
  int m16  = lane & 15;
  int kh   = lane >> 4;                 // lane half selects K pair / M half
  const size_t plane = (size_t)LEN * CH;
  const float* xb = x + (size_t)b * plane;
  const float* mb = mm + (size_t)b * plane;

  for (int i = tid; i < KSEL * 16; i += 256) ((float*)c2)[i] = 0.f;
  __syncthreads();

  // ---- phase A: C[k][ch] = sum_n Bt[k][n] * (x - mm)[n][ch] ----
  float* bt_w = ldsbuf + w * (80 * 33);
  v8f acc[5] = {};
  for (int chunk = 0; chunk < 16; ++chunk) {
    int nbase = w * 512 + chunk * 32;
    __syncthreads();
    for (int idx = lane; idx < 80 * 32; idx += 32) {
      int k = idx >> 5, nn = idx & 31;
      bt_w[k * 33 + nn] = Bt[(size_t)k * 4096 + nbase + nn];
    }
    __syncthreads();
    #pragma unroll
    for (int ks = 0; ks < 8; ++ks) {
      int nn = ks * 4;
      int nrow = nbase + nn + 2 * kh;
      size_t a0 = (size_t)nrow * CH + c0 + m16;
      v2f Bop;                           // B(4x16): rows 2kh,2kh+1, col m16
      Bop.x = xb[a0] - mb[a0];
      Bop.y = xb[a0 + CH] - mb[a0 + CH];
      #pragma unroll
      for (int t = 0; t < 5; ++t) {      // 5 M-tiles cover 80 k-rows
        v2f Aop;                         // A(16x4): row m16, cols nn+2kh,+1
        Aop.x = bt_w[(t * 16 + m16) * 33 + nn + 2 * kh];
        Aop.y = bt_w[(t * 16 + m16) * 33 + nn + 2 * kh + 1];
        acc[t] = __builtin_amdgcn_wmma_f32_16x16x4_f32(false, Aop, false, Bop,
                                                       (short)0, acc[t], false, false);
      }
    }
  }
  #pragma unroll
  for (int t = 0; t < 5; ++t) {
    #pragma unroll
    for (int r = 0; r < 8; ++r) {        // D: VGPR r -> M = r + 8*kh, N = m16
      atomicAdd(&c2[t * 16 + r + 8 * kh][m16], acc[t][r]);
    }
  }
  __syncthreads();
  for (int i = tid; i < KSEL * 16; i += 256) {
    int k = i >> 4, ch = i & 15;
    c2[k][ch] *= coef[k >> 1];
  }
  __syncthreads();

  // ---- phase B: out[n][ch] = 0.9*s + sum_k B1[n][k]*c2[k][ch] ----
  float* b1_w = ldsbuf + w * (16 * 81);
  float* ob = outF + (size_t)b * plane;
  for (int i = 0; i < 32; ++i) {
    int n0 = (w + 8 * i) * 16;
    __syncthreads();
    for (int idx = lane; idx < 16 * 80; idx += 32) {
      int r = idx / 80, k = idx % 80;
      b1_w[r * 81 + k] = B1[(size_t)(n0 + r) * KSEL + k];
    }
    __syncthreads();
    v8f d = {};
    #pragma unroll
    for (int ks = 0; ks < 20; ++ks) {
      int kk = ks * 4;
      v2f Aop, Bop;
      Aop.x = b1_w[m16 * 81 + kk + 2 * kh];
      Aop.y = b1_w[m16 * 81 + kk + 2 * kh + 1];
      Bop.x = c2[kk + 2 * kh][m16];
      Bop.y = c2[kk + 2 * kh + 1][m16];
      d = __builtin_amdgcn_wmma_f32_16x16x4_f32(false, Aop, false, Bop,
                                                (short)0, d, false, false);
    }
    #pragma unroll
    for (int r = 0; r < 8; ++r) {
      int n = n0 + r + 8 * kh;
      size_t a = (size_t)n * CH + c0 + m16;
      float sv = xb[a] - mb[a];
      ob[a] = 0.9f * sv + d[r];
    }
  }
}

// ---------------- launcher ----------------
extern "C" void kernel_launch(void* const* d_in, const int* in_sizes, int n_in,
                              void* d_out, int out_size, void* d_ws, size_t ws_size,
                              hipStream_t stream) {
  (void)in_sizes; (void)n_in; (void)out_size; (void)ws_size;
  const float* x = (const float*)d_in[0];
  float* outF = (float*)d_out;                          // final_seasonal
  float* outM = outF + (size_t)BATCH * LEN * CH;        // moving_mean
  float* ws   = (float*)d_ws;
  float* st   = ws + ST_OFF;
  float* ap   = ws + AP_OFF;
  int*   fsel = (int*)(ws + FSEL_OFF);
  float* coef = ws + COEF_OFF;
  float* B1   = ws + B1_OFF;
  float* Bt   = ws + BT_OFF;

  hipLaunchKernelGGL(k0_init, dim3(1), dim3(256), 0, stream, ap);
  hipLaunchKernelGGL(kA_decomp, dim3(CH / 32, LEN / 32, BATCH), dim3(256), 0, stream,
                     x, outM, st);
  hipLaunchKernelGGL(kB_fftpow, dim3(1024), dim3(256), 0, stream, st, ap);
  hipLaunchKernelGGL(kC_mask, dim3(1), dim3(256), 0, stream, ap, fsel, coef, B1, Bt);
  hipLaunchKernelGGL(kD_final, dim3(CH / 16, BATCH), dim3(256), 0, stream,
                     x, outM, B1, Bt, coef, outF);
}